// GCNSyntheticPerturbEdgeWeight_75539884802086
// MI455X (gfx1250) — compile-verified
//
#include <hip/hip_runtime.h>
#include <hip/hip_bf16.h>

typedef __attribute__((ext_vector_type(2))) float v2f;
typedef __attribute__((ext_vector_type(8))) float v8f;

// ---------------------------------------------------------------------------
// Edge preprocessing
// ---------------------------------------------------------------------------

// ew[e] = sigmoid(ewp[e]);  deg[dst[e]] += ew[e]   (deg pre-zeroed)
__global__ void deg_kernel(const float* __restrict__ ewp, const int* __restrict__ dst,
                           float* __restrict__ deg, float* __restrict__ ew, int E) {
    int e = blockIdx.x * blockDim.x + threadIdx.x;
    if (e >= E) return;
    float w = 1.0f / (1.0f + __expf(-ewp[e]));
    ew[e] = w;
    atomicAdd(&deg[dst[e]], w);
}

// deg -> dinv in place, including self-loop weight 1
__global__ void dinv_kernel(float* __restrict__ deg, int N) {
    int n = blockIdx.x * blockDim.x + threadIdx.x;
    if (n >= N) return;
    float d = deg[n] + 1.0f;
    deg[n] = (d > 0.0f) ? rsqrtf(fmaxf(d, 1e-12f)) : 0.0f;
}

// norm[e] = dinv[src]*ew*dinv[dst]
__global__ void norm_kernel(const int* __restrict__ src, const int* __restrict__ dst,
                            const float* __restrict__ ew, const float* __restrict__ dinv,
                            float* __restrict__ norm, int E) {
    int e = blockIdx.x * blockDim.x + threadIdx.x;
    if (e >= E) return;
    norm[e] = dinv[src[e]] * ew[e] * dinv[dst[e]];
}

// ---------------------------------------------------------------------------
// Weight pre-pack into WMMA B-fragment order (zero-padded to NTILES*16 cols).
// Fragment for lane l at k-step ks, tile t:
//   half = l>>4, col = t*16 + (l&15), ka = ks*4 + half*2
//   values { W[ka][col], W[ka+1][col] }  stored contiguously:
//   Wp[ ((ks*NTILES + t)*32 + l) * 2 + {0,1} ]
// ---------------------------------------------------------------------------
template<int KDIM, int NTILES>
__global__ void pack_b_kernel(const float* __restrict__ W, int Nout, float* __restrict__ Wp) {
    constexpr int KSTEPS = KDIM / 4;
    int i = blockIdx.x * blockDim.x + threadIdx.x;       // i = (ks*NTILES+t)*32 + lane
    if (i >= KSTEPS * NTILES * 32) return;
    int lane = i & 31;
    int frag = i >> 5;
    int t  = frag % NTILES;
    int ks = frag / NTILES;
    int half = lane >> 4, l16 = lane & 15;
    int col = t * 16 + l16;
    int ka  = ks * 4 + half * 2;
    float b0 = (col < Nout) ? W[ka * Nout + col]       : 0.0f;
    float b1 = (col < Nout) ? W[(ka + 1) * Nout + col] : 0.0f;
    Wp[(size_t)i * 2 + 0] = b0;
    Wp[(size_t)i * 2 + 1] = b1;
}

// pad bias to 16 entries (zeros beyond Nout)
__global__ void pad_bias_kernel(const float* __restrict__ b, float* __restrict__ bp, int Nout, int P) {
    int i = threadIdx.x;
    if (i < P) bp[i] = (i < Nout) ? b[i] : 0.0f;
}

// ---------------------------------------------------------------------------
// Dense GEMM via V_WMMA_F32_16X16X4_F32 (fp32 in / fp32 acc — matches ref).
// C[M x NTILES*16] = A[M x KDIM] @ Wp (pre-packed fragments), row-major.
// One wave per 16-row tile; blockDim 128 = 4 waves. Wave-uniform early exit
// keeps EXEC all-ones at every WMMA. Inner loop: 1+NTILES b64 loads + WMMAs.
// ---------------------------------------------------------------------------
template<int KDIM, int NTILES>
__global__ void gemm_wmma_f32(const float* __restrict__ A, const float* __restrict__ Wp,
                              float* __restrict__ C, int M) {
    constexpr int KSTEPS = KDIM / 4;
    constexpr int NOUT   = NTILES * 16;
    int lane  = threadIdx.x & 31;
    int wave  = threadIdx.x >> 5;
    int mtile = blockIdx.x * 4 + wave;
    int MT    = (M + 15) >> 4;
    if (mtile >= MT) return;                       // uniform across the wave

    int m0   = mtile * 16;
    int half = lane >> 4;
    int l16  = lane & 15;
    int row  = m0 + l16;
    if (row >= M) row = M - 1;                     // no-op when M % 16 == 0

    const float* arow = A + (size_t)row * KDIM + half * 2;
    const v2f*   wp   = (const v2f*)Wp + lane;     // + (ks*NTILES+t)*32 per fragment

    v8f acc[NTILES] = {};

    for (int ks = 0; ks < KSTEPS; ++ks) {
        v2f a = *(const v2f*)(arow + ks * 4);      // b64, 8B-aligned
#pragma unroll
        for (int t = 0; t < NTILES; ++t) {
            v2f b = wp[(ks * NTILES + t) * 32];    // b64, fragment-packed
            acc[t] = __builtin_amdgcn_wmma_f32_16x16x4_f32(
                false, a, false, b, (short)0, acc[t], false, false);
        }
    }

    float* crow = C + (size_t)m0 * NOUT;
#pragma unroll
    for (int t = 0; t < NTILES; ++t) {
        int col = t * 16 + l16;
#pragma unroll
        for (int v = 0; v < 8; ++v) {
            int r = v + half * 8;                  // C/D layout: vgpr v, lane-half
            if (m0 + r < M) crow[(size_t)r * NOUT + col] = acc[t][v];
        }
    }
}

// ---------------------------------------------------------------------------
// Edge scatter: agg[dst] += h[src] * norm  (plus self-loops with dinv^2).
// One thread per (virtual edge, 4 features); agg stays L2-resident.
// ---------------------------------------------------------------------------
template<int D>
__global__ void scatter_kernel(const float* __restrict__ h, const int* __restrict__ src,
                               const int* __restrict__ dst, const float* __restrict__ norm,
                               const float* __restrict__ dinv, float* __restrict__ agg,
                               int E, int N) {
    constexpr int DV = D / 4;
    long long idx = (long long)blockIdx.x * blockDim.x + threadIdx.x;
    long long total = (long long)(E + N) * DV;
    if (idx >= total) return;
    int e  = (int)(idx / DV);
    int f4 = (int)(idx % DV) * 4;
    int s, d; float w;
    if (e < E) {
        s = src[e]; d = dst[e]; w = norm[e];
    } else {
        s = d = e - E;
        float dv = dinv[s];
        w = dv * dv;
    }
    float4 hv = *(const float4*)(h + (size_t)s * D + f4);
    float* a = agg + (size_t)d * D + f4;
    atomicAdd(a + 0, hv.x * w);
    atomicAdd(a + 1, hv.y * w);
    atomicAdd(a + 2, hv.z * w);
    atomicAdd(a + 3, hv.w * w);
}

// agg = (relu?) (agg + b),  D a power of two
template<int D>
__global__ void bias_act_kernel(float* __restrict__ agg, const float* __restrict__ b,
                                int N, int do_relu) {
    long long idx = (long long)blockIdx.x * blockDim.x + threadIdx.x;
    if (idx >= (long long)N * D) return;
    int f = (int)(idx & (D - 1));
    float v = agg[idx] + b[f];
    agg[idx] = do_relu ? fmaxf(v, 0.0f) : v;
}

// log_softmax over one row (stride-S rows, first D entries); single wave32
__global__ void logsoftmax_kernel(const float* __restrict__ h, const int* __restrict__ idxp,
                                  float* __restrict__ out, int D, int S) {
    int row  = idxp[0];
    int lane = threadIdx.x;
    const float* r = h + (long long)row * S;
    float v = (lane < D) ? r[lane] : -1e30f;
    float m = v;
#pragma unroll
    for (int off = 16; off > 0; off >>= 1) m = fmaxf(m, __shfl_xor(m, off));
    float ex = (lane < D) ? __expf(v - m) : 0.0f;
    float s = ex;
#pragma unroll
    for (int off = 16; off > 0; off >>= 1) s += __shfl_xor(s, off);
    if (lane < D) out[lane] = v - m - __logf(s);
}

// ---------------------------------------------------------------------------
// Launch
// ---------------------------------------------------------------------------
extern "C" void kernel_launch(void* const* d_in, const int* in_sizes, int n_in,
                              void* d_out, int out_size, void* d_ws, size_t ws_size,
                              hipStream_t stream) {
    const float* x   = (const float*)d_in[0];
    const int*   ei  = (const int*)  d_in[1];
    const float* ewp = (const float*)d_in[2];
    const float* W1  = (const float*)d_in[3];
    const float* b1  = (const float*)d_in[4];
    const float* W2  = (const float*)d_in[5];
    const float* b2  = (const float*)d_in[6];
    const float* W3  = (const float*)d_in[7];
    const float* b3  = (const float*)d_in[8];
    const int*   idx = (const int*)  d_in[9];
    float* out = (float*)d_out;

    const int DIN = 128, DH = 64, DC = 10, DCP = 16;
    const int N = in_sizes[0] / DIN;
    const int E = in_sizes[1] / 2;
    const int* src = ei;         // edge_index[0]
    const int* dst = ei + E;     // edge_index[1]

    // workspace layout (floats)
    float* ws   = (float*)d_ws;
    float* ew   = ws;                          // [E]
    float* norm = ew + E;                      // [E]
    float* dinv = norm + E;                    // [N]
    float* t    = dinv + N;                    // [N*64]
    float* agg  = t + (size_t)N * DH;          // [N*64]
    float* Wp1  = agg + (size_t)N * DH;        // [32*4*32*2] = 8192
    float* Wp2  = Wp1 + 8192;                  // [16*4*32*2] = 4096
    float* Wp3  = Wp2 + 4096;                  // [16*1*32*2] = 1024
    float* b3p  = Wp3 + 1024;                  // [16]

    const int TB = 256;

    // --- weight packing (weights constant; cheap, L2-resident) ---
    pack_b_kernel<128, 4><<<(32 * 4 * 32 + TB - 1) / TB, TB, 0, stream>>>(W1, DH, Wp1);
    pack_b_kernel< 64, 4><<<(16 * 4 * 32 + TB - 1) / TB, TB, 0, stream>>>(W2, DH, Wp2);
    pack_b_kernel< 64, 1><<<(16 * 1 * 32 + TB - 1) / TB, TB, 0, stream>>>(W3, DC, Wp3);
    pad_bias_kernel<<<1, 32, 0, stream>>>(b3, b3p, DC, DCP);

    // --- normalization ---
    hipMemsetAsync(dinv, 0, (size_t)N * sizeof(float), stream);
    deg_kernel <<<(E + TB - 1) / TB, TB, 0, stream>>>(ewp, dst, dinv, ew, E);
    dinv_kernel<<<(N + TB - 1) / TB, TB, 0, stream>>>(dinv, N);
    norm_kernel<<<(E + TB - 1) / TB, TB, 0, stream>>>(src, dst, ew, dinv, norm, E);

    const int MT = (N + 15) / 16;
    const int gemmBlocks = (MT + 3) / 4;       // 4 waves (128 threads) per block
    const long long sH = (long long)(E + N) * (DH / 4);
    const long long sC = (long long)(E + N) * (DCP / 4);

    // --- layer 1 ---
    gemm_wmma_f32<128, 4><<<gemmBlocks, 128, 0, stream>>>(x, Wp1, t, N);
    hipMemsetAsync(agg, 0, (size_t)N * DH * sizeof(float), stream);
    scatter_kernel<64><<<(int)((sH + TB - 1) / TB), TB, 0, stream>>>(t, src, dst, norm, dinv, agg, E, N);
    bias_act_kernel<64><<<(int)(((long long)N * DH + TB - 1) / TB), TB, 0, stream>>>(agg, b1, N, 1);

    // --- layer 2 ---
    gemm_wmma_f32<64, 4><<<gemmBlocks, 128, 0, stream>>>(agg, Wp2, t, N);
    hipMemsetAsync(agg, 0, (size_t)N * DH * sizeof(float), stream);
    scatter_kernel<64><<<(int)((sH + TB - 1) / TB), TB, 0, stream>>>(t, src, dst, norm, dinv, agg, E, N);
    bias_act_kernel<64><<<(int)(((long long)N * DH + TB - 1) / TB), TB, 0, stream>>>(agg, b2, N, 1);

    // --- layer 3 (padded to 16 cols; pads are exact zeros; no relu) ---
    gemm_wmma_f32<64, 1><<<gemmBlocks, 128, 0, stream>>>(agg, Wp3, t, N);
    hipMemsetAsync(agg, 0, (size_t)N * DCP * sizeof(float), stream);
    scatter_kernel<16><<<(int)((sC + TB - 1) / TB), TB, 0, stream>>>(t, src, dst, norm, dinv, agg, E, N);
    bias_act_kernel<16><<<(int)(((long long)N * DCP + TB - 1) / TB), TB, 0, stream>>>(agg, b3p, N, 0);

    // --- output: log_softmax(row `index`, first 10 of stride-16 row) ---
    logsoftmax_kernel<<<1, 32, 0, stream>>>(agg, idx, out, DC, DCP);
}